// ProDIGY_78357383348249
// MI455X (gfx1250) — compile-verified
//
#include <hip/hip_runtime.h>
#include <hip/hip_bf16.h>
#include <math.h>

#define NUM_NODES   50000
#define NUM_GRAPHS  1024
#define NUM_CLASSES 8
#define DIM         128
#define TASK_N      (NUM_GRAPHS + NUM_CLASSES)   // 1032

typedef float v2f __attribute__((ext_vector_type(2)));
typedef float v8f __attribute__((ext_vector_type(8)));

// ---------------------------------------------------------------------------
// WMMA f32 GEMM:  C[M x 128] = A[M x 128] * B[128 x 128]
// One wave32 per 16-row tile; the wave produces all 8 column tiles (N=128).
// A-frag (16x4 f32): lanes 0-15 hold K=k,k+1 of row M=lane; lanes 16-31 hold
// K=k+2,k+3 of row M=lane-16.  B-frag (4x16): VGPR0 = rows K=k (lanes 0-15) /
// K=k+2 (lanes 16-31), VGPR1 = K=k+1 / K=k+3, N = lane%16.
// C (16x16 f32): VGPR r -> row r (lanes 0-15) / row r+8 (lanes 16-31), N=lane%16.
// ---------------------------------------------------------------------------
__global__ void gemm128_wmma(const float* __restrict__ A,
                             const float* __restrict__ B,
                             float* __restrict__ C, int M) {
    int wave = (blockIdx.x * blockDim.x + threadIdx.x) >> 5;
    int lane = threadIdx.x & 31;
    int row0 = wave * 16;
    if (row0 >= M) return;

    int half = lane >> 4;        // 0: lanes 0-15, 1: lanes 16-31
    int l16  = lane & 15;

    int arow = row0 + l16;
    if (arow > M - 1) arow = M - 1;          // branchless clamp (EXEC stays full)
    const float* arowp = A + (size_t)arow * DIM;

    v8f acc[8];
#pragma unroll
    for (int n = 0; n < 8; ++n) acc[n] = (v8f)0.0f;

#pragma unroll 2
    for (int k = 0; k < DIM; k += 4) {
        int ka = k + 2 * half;
        v2f a;
        a.x = arowp[ka + 0];
        a.y = arowp[ka + 1];
#pragma unroll
        for (int n = 0; n < 8; ++n) {
            int col = n * 16 + l16;
            v2f b;
            b.x = B[(size_t)(ka + 0) * DIM + col];
            b.y = B[(size_t)(ka + 1) * DIM + col];
            acc[n] = __builtin_amdgcn_wmma_f32_16x16x4_f32(
                false, a, false, b, (short)0, acc[n], false, false);
        }
    }

    int crow0 = row0 + half * 8;
#pragma unroll
    for (int r = 0; r < 8; ++r) {
        int crow = crow0 + r;
        if (crow < M) {
            float* cp = C + (size_t)crow * DIM + l16;
#pragma unroll
            for (int n = 0; n < 8; ++n) cp[n * 16] = acc[n][r];
        }
    }
}

// ---------------------------------------------------------------------------
// small elementwise / init kernels
// ---------------------------------------------------------------------------
__global__ void init_const(float* p, float v, int n) {
    int i = blockIdx.x * blockDim.x + threadIdx.x;
    if (i < n) p[i] = v;
}

__global__ void deg_edges(const int* __restrict__ ei, float* deg, int E) {
    int e = blockIdx.x * blockDim.x + threadIdx.x;
    if (e < E) {
        atomicAdd(&deg[ei[e]], 1.0f);        // node as dst of reversed edge
        atomicAdd(&deg[ei[E + e]], 1.0f);    // node as dst of forward edge
    }
}

__global__ void rsqrt_inplace(float* p, int n) {
    int i = blockIdx.x * blockDim.x + threadIdx.x;
    if (i < n) { float d = p[i]; p[i] = d > 0.0f ? rsqrtf(d) : 0.0f; }
}

// acc[i,:] = h[i,:] * dinv[i]^2   (self-loop contribution)
__global__ void self_init(const float* __restrict__ h, const float* __restrict__ dinv,
                          float* __restrict__ acc, int n_nodes) {
    int i = blockIdx.x * blockDim.x + threadIdx.x;
    if (i < n_nodes * DIM) {
        float w = dinv[i >> 7];
        acc[i] = h[i] * w * w;
    }
}

// One wave per directed edge (both directions of the undirected graph).
// segment ids fed to segment_sum: s = concat(src, dst), d = concat(dst, src).
// Edge endpoints are wave-uniform -> keep them scalar via readfirstlane.
__global__ void edge_agg(const float* __restrict__ h, const int* __restrict__ ei,
                         const float* __restrict__ dinv, float* __restrict__ acc, int E) {
    int wid  = (blockIdx.x * blockDim.x + threadIdx.x) >> 5;
    int lane = threadIdx.x & 31;
    if (wid >= 2 * E) return;
    int s, d;
    if (wid < E) { s = ei[wid];     d = ei[E + wid]; }   // forward edge
    else         { int e = wid - E; s = ei[E + e]; d = ei[e]; }  // mirrored edge
    s = __builtin_amdgcn_readfirstlane(s);
    d = __builtin_amdgcn_readfirstlane(d);
    float nrm = dinv[s] * dinv[d];
    float4 v = ((const float4*)(h + (size_t)s * DIM))[lane];
    float* ap = acc + (size_t)d * DIM + lane * 4;
    atomicAdd(ap + 0, v.x * nrm);
    atomicAdd(ap + 1, v.y * nrm);
    atomicAdd(ap + 2, v.z * nrm);
    atomicAdd(ap + 3, v.w * nrm);
}

__global__ void bias_relu(float* __restrict__ p, const float* __restrict__ b, int n_nodes) {
    int i = blockIdx.x * blockDim.x + threadIdx.x;
    if (i < n_nodes * DIM) {
        float v = p[i] + b[i & (DIM - 1)];
        p[i] = v > 0.0f ? v : 0.0f;
    }
}

// full_x: first 1024*128 zeroed (pool target), class_nodes appended
__global__ void fullx_init(const float* __restrict__ cls, float* __restrict__ fx) {
    int i = blockIdx.x * blockDim.x + threadIdx.x;
    if (i < TASK_N * DIM)
        fx[i] = (i < NUM_GRAPHS * DIM) ? 0.0f : cls[i - NUM_GRAPHS * DIM];
}

// global_add_pool with +b2 folded in; one wave per node
__global__ void pool_nodes(const float* __restrict__ acc2, const float* __restrict__ b2,
                           const int* __restrict__ batch, float* __restrict__ gx) {
    int wid  = (blockIdx.x * blockDim.x + threadIdx.x) >> 5;
    int lane = threadIdx.x & 31;
    if (wid >= NUM_NODES) return;
    int g = __builtin_amdgcn_readfirstlane(batch[wid]);
    float4 v  = ((const float4*)(acc2 + (size_t)wid * DIM))[lane];
    float4 bb = ((const float4*)b2)[lane];
    float* dst = gx + (size_t)g * DIM + lane * 4;
    atomicAdd(dst + 0, v.x + bb.x);
    atomicAdd(dst + 1, v.y + bb.y);
    atomicAdd(dst + 2, v.z + bb.z);
    atomicAdd(dst + 3, v.w + bb.w);
}

__global__ void task_deg(const int* __restrict__ lab, float* tdeg) {
    int e = blockIdx.x * blockDim.x + threadIdx.x;
    if (e < NUM_GRAPHS) {
        atomicAdd(&tdeg[lab[e]], 1.0f);   // forward edge: dst = label node index
        atomicAdd(&tdeg[e], 1.0f);        // reversed edge: dst = graph node
    }
}

__global__ void task_edge_agg(const float* __restrict__ h, const int* __restrict__ lab,
                              const float* __restrict__ dinv, float* __restrict__ acc) {
    int wid  = (blockIdx.x * blockDim.x + threadIdx.x) >> 5;
    int lane = threadIdx.x & 31;
    if (wid >= 2 * NUM_GRAPHS) return;
    int s, d;
    if (wid < NUM_GRAPHS) { s = wid;                    d = lab[wid]; }
    else                  { s = lab[wid - NUM_GRAPHS];  d = wid - NUM_GRAPHS; }
    s = __builtin_amdgcn_readfirstlane(s);
    d = __builtin_amdgcn_readfirstlane(d);
    float nrm = dinv[s] * dinv[d];
    float4 v = ((const float4*)(h + (size_t)s * DIM))[lane];
    float* ap = acc + (size_t)d * DIM + lane * 4;
    atomicAdd(ap + 0, v.x * nrm);
    atomicAdd(ap + 1, v.y * nrm);
    atomicAdd(ap + 2, v.z * nrm);
    atomicAdd(ap + 3, v.w * nrm);
}

// logits = (acct[row] + bt) @ Wfc + bfc ; log_softmax over 8 classes
__global__ void fc_logsoftmax(const float* __restrict__ acct, const float* __restrict__ bt,
                              const float* __restrict__ Wfc, const float* __restrict__ bfc,
                              float* __restrict__ out) {
    int row = blockIdx.x * blockDim.x + threadIdx.x;
    if (row >= NUM_GRAPHS) return;
    const float* a = acct + (size_t)row * DIM;
    float lg[NUM_CLASSES];
    float mx = -INFINITY;
#pragma unroll
    for (int j = 0; j < NUM_CLASSES; ++j) {
        float s = bfc[j];
        for (int k = 0; k < DIM; ++k) s += (a[k] + bt[k]) * Wfc[k * NUM_CLASSES + j];
        lg[j] = s;
        mx = fmaxf(mx, s);
    }
    float se = 0.0f;
#pragma unroll
    for (int j = 0; j < NUM_CLASSES; ++j) se += expf(lg[j] - mx);
    float l = mx + logf(se);
#pragma unroll
    for (int j = 0; j < NUM_CLASSES; ++j) out[row * NUM_CLASSES + j] = lg[j] - l;
}

// ---------------------------------------------------------------------------
extern "C" void kernel_launch(void* const* d_in, const int* in_sizes, int n_in,
                              void* d_out, int out_size, void* d_ws, size_t ws_size,
                              hipStream_t stream) {
    const float* x    = (const float*)d_in[0];
    const int*   ei   = (const int*)  d_in[1];
    const int*   batch= (const int*)  d_in[2];
    const int*   slab = (const int*)  d_in[3];
    const float* W1   = (const float*)d_in[4];
    const float* b1   = (const float*)d_in[5];
    const float* W2   = (const float*)d_in[6];
    const float* b2   = (const float*)d_in[7];
    const float* Wt   = (const float*)d_in[8];
    const float* bt   = (const float*)d_in[9];
    const float* cls  = (const float*)d_in[10];
    const float* Wfc  = (const float*)d_in[11];
    const float* bfc  = (const float*)d_in[12];
    float* out = (float*)d_out;

    const int E = in_sizes[1] / 2;               // 600000 directed edges

    float* ws    = (float*)d_ws;
    float* dinv  = ws;                                   // 50048
    float* bufA  = dinv + 50048;                         // 6.4M  (GEMM out / msg src)
    float* bufB  = bufA + (size_t)NUM_NODES * DIM;       // 6.4M  (aggregation acc)
    float* fullx = bufB + (size_t)NUM_NODES * DIM;       // 1032*128
    float* htp   = fullx + TASK_N * DIM;                 // 1032*128
    float* acct  = htp   + TASK_N * DIM;                 // 1032*128
    float* tdinv = acct  + TASK_N * DIM;                 // 1056

    const int NB = 256;
    const int nNodeElems = NUM_NODES * DIM;

    // ---- degrees (self-loop baked in as init 1.0) -> dinv ----
    init_const<<<(NUM_NODES + NB - 1) / NB, NB, 0, stream>>>(dinv, 1.0f, NUM_NODES);
    deg_edges <<<(E + NB - 1) / NB, NB, 0, stream>>>(ei, dinv, E);
    rsqrt_inplace<<<(NUM_NODES + NB - 1) / NB, NB, 0, stream>>>(dinv, NUM_NODES);

    // ---- layer 1: h = x @ W1 ; aggregate ; +b1 ; relu ----
    {
        int waves = (NUM_NODES + 15) / 16;
        gemm128_wmma<<<(waves + 3) / 4, 128, 0, stream>>>(x, W1, bufA, NUM_NODES);
    }
    self_init<<<(nNodeElems + NB - 1) / NB, NB, 0, stream>>>(bufA, dinv, bufB, NUM_NODES);
    edge_agg <<<(2 * E * 32 + NB - 1) / NB, NB, 0, stream>>>(bufA, ei, dinv, bufB, E);
    bias_relu<<<(nNodeElems + NB - 1) / NB, NB, 0, stream>>>(bufB, b1, NUM_NODES);

    // ---- layer 2: h2 = h1r @ W2 ; aggregate ----
    {
        int waves = (NUM_NODES + 15) / 16;
        gemm128_wmma<<<(waves + 3) / 4, 128, 0, stream>>>(bufB, W2, bufA, NUM_NODES);
    }
    self_init<<<(nNodeElems + NB - 1) / NB, NB, 0, stream>>>(bufA, dinv, bufB, NUM_NODES);
    edge_agg <<<(2 * E * 32 + NB - 1) / NB, NB, 0, stream>>>(bufA, ei, dinv, bufB, E);

    // ---- pool (+b2) into full_x[0:1024], append class nodes ----
    fullx_init<<<(TASK_N * DIM + NB - 1) / NB, NB, 0, stream>>>(cls, fullx);
    pool_nodes<<<(NUM_NODES * 32 + NB - 1) / NB, NB, 0, stream>>>(bufB, b2, batch, fullx);

    // ---- task graph GCN over 1032 nodes ----
    init_const<<<(TASK_N + NB - 1) / NB, NB, 0, stream>>>(tdinv, 1.0f, TASK_N);
    task_deg  <<<(NUM_GRAPHS + NB - 1) / NB, NB, 0, stream>>>(slab, tdinv);
    rsqrt_inplace<<<(TASK_N + NB - 1) / NB, NB, 0, stream>>>(tdinv, TASK_N);
    {
        int waves = (TASK_N + 15) / 16;   // 65 tiles, last one ragged (clamped/guarded)
        gemm128_wmma<<<(waves + 3) / 4, 128, 0, stream>>>(fullx, Wt, htp, TASK_N);
    }
    self_init<<<(TASK_N * DIM + NB - 1) / NB, NB, 0, stream>>>(htp, tdinv, acct, TASK_N);
    task_edge_agg<<<(2 * NUM_GRAPHS * 32 + NB - 1) / NB, NB, 0, stream>>>(htp, slab, tdinv, acct);

    // ---- FC (+bt folded) + log_softmax -> d_out [1024 x 8] ----
    fc_logsoftmax<<<(NUM_GRAPHS + 127) / 128, 128, 0, stream>>>(acct, bt, Wfc, bfc, out);
}